// AggrSum_84043920048428
// MI455X (gfx1250) — compile-verified
//
#include <hip/hip_runtime.h>
#include <stdint.h>

// ---------------------------------------------------------------------------
// Segment-sum (scatter-add) for MI455X / gfx1250.
//   traffic: 204.8MB H + 6.4MB idx + 12.8MB out RMW  ->  ~10us @ 23.3 TB/s
//   flops:   51.2M adds (irrelevant) -> memory-bound; WMMA is the wrong tool
// CDNA5 path: Tensor Data Mover streams 16KB H tiles global->LDS
// (tensor_load_to_lds + s_wait_tensorcnt, double-buffered), 8 waves drain LDS
// into native f32 atomics; the 12.8MB output lives in the 192MB L2.
// ---------------------------------------------------------------------------

#define TILE_EDGES  64
#define DFEAT       64
#define TILE_FLOATS (TILE_EDGES * DFEAT)   // 4096 floats = 16KB per tile
#define BLOCK       256

typedef __attribute__((ext_vector_type(4))) unsigned tdm_g0_t;
typedef __attribute__((ext_vector_type(8))) unsigned tdm_g1_t;

// 1-D TDM copy of `nelem` f32 elements from gsrc into LDS at byte offset
// lds_off, 2-group (<=2D) descriptor form:
//   group0: [1:0]=count(1), [63:32]=lds_addr, [120:64]=global_addr, [127:126]=type(2)
//   group1: [17:16]=data_size(2->4B), [79:48]=tensor_dim0, [127:112]=tile_dim0,
//           [207:160]=tensor_dim0_stride
// tensor_dim0 = nelem gives free zero-fill past the tensor end.
__device__ __forceinline__ void tdm_load_f32_1d(const float* gsrc,
                                                unsigned lds_off,
                                                unsigned nelem) {
  unsigned long long ga = (unsigned long long)(uintptr_t)gsrc;
  tdm_g0_t g0;
  g0.x = 1u;                                                // count=1, user desc
  g0.y = lds_off;                                           // lds_addr
  g0.z = (unsigned)ga;                                      // global_addr[31:0]
  g0.w = ((unsigned)(ga >> 32) & 0x01FFFFFFu) | (2u << 30); // ga[56:32] | type=2
  tdm_g1_t g1;
  g1.s0 = 2u << 16;                    // workgroup_mask=0, data_size=2 (4B)
  g1.s1 = (nelem & 0xFFFFu) << 16;     // tensor_dim0[15:0]  @ bits[63:48]
  g1.s2 = (nelem >> 16) & 0xFFFFu;     // tensor_dim0[31:16] @ bits[79:64]
  g1.s3 = (nelem & 0xFFFFu) << 16;     // tile_dim0 @ bits[127:112] (<=65535)
  g1.s4 = 0u;                          // tile_dim1=0, tile_dim2=0 (unused)
  g1.s5 = nelem;                       // tensor_dim0_stride[31:0]
  g1.s6 = 0u;
  g1.s7 = 0u;
  asm volatile("tensor_load_to_lds %0, %1" :: "s"(g0), "s"(g1) : "memory");
}

__global__ void __launch_bounds__(BLOCK)
aggr_scatter_kernel(const float* __restrict__ H,
                    const long long* __restrict__ Xn,
                    float* __restrict__ out,
                    int E, int nTiles) {
  __shared__ float hbuf[2][TILE_FLOATS];   // 32 KB double buffer
  __shared__ int   ibuf[2][TILE_EDGES];    // staged (narrowed) indices

  const int  tid    = threadIdx.x;
  const bool leader = (tid < 32);          // wave 0 owns TDM issue + wait

  int tile = blockIdx.x;
  if (tile >= nTiles) return;

  // ---- prologue: stage first tile into buffer 0 ----
  if (leader) {
    int base = tile * TILE_FLOATS;
    int nel  = min(TILE_FLOATS, E * DFEAT - base);
    tdm_load_f32_1d(H + base, (unsigned)(uintptr_t)&hbuf[0][0], (unsigned)nel);
  }
  if (tid < TILE_EDGES) {
    int e = tile * TILE_EDGES + tid;
    ibuf[0][tid] = (e < E) ? (int)Xn[e] : 0;
  }

  int cur = 0;
  for (; tile < nTiles; tile += gridDim.x, cur ^= 1) {
    const int  next    = tile + gridDim.x;
    const bool hasNext = next < nTiles;

    // ---- prefetch next tile into the other buffer ----
    if (hasNext) {
      if (leader) {
        int base = next * TILE_FLOATS;
        int nel  = min(TILE_FLOATS, E * DFEAT - base);
        tdm_load_f32_1d(H + base, (unsigned)(uintptr_t)&hbuf[cur ^ 1][0],
                        (unsigned)nel);
      }
      if (tid < TILE_EDGES) {
        int e = next * TILE_EDGES + tid;
        ibuf[cur ^ 1][tid] = (e < E) ? (int)Xn[e] : 0;
      }
    }

    // ---- retire the current tile's DMA (TENSORcnt is in-order) ----
    if (leader) {
      if (hasNext) __builtin_amdgcn_s_wait_tensorcnt(1); // newest may still fly
      else         __builtin_amdgcn_s_wait_tensorcnt(0);
    }
    __syncthreads();   // publish hbuf[cur] + ibuf[cur] to all 8 waves

    // ---- scatter: lanes read consecutive LDS dwords (bank-conflict-free),
    //      each wave's 32 atomics cover one 128B half-row of `out` ----
    const int tileBase = tile * TILE_EDGES;
    if (tileBase + TILE_EDGES <= E) {
      // full tile: no per-element guards
#pragma unroll
      for (int k = 0; k < TILE_FLOATS / BLOCK; ++k) {
        int j = tid + k * BLOCK;
        int n = ibuf[cur][j >> 6];
        unsafeAtomicAdd(&out[(size_t)n * DFEAT + (j & (DFEAT - 1))],
                        hbuf[cur][j]);
      }
    } else {
      // tail tile
#pragma unroll
      for (int k = 0; k < TILE_FLOATS / BLOCK; ++k) {
        int j = tid + k * BLOCK;
        int e = j >> 6;
        if (tileBase + e < E) {
          int n = ibuf[cur][e];
          unsafeAtomicAdd(&out[(size_t)n * DFEAT + (j & (DFEAT - 1))],
                          hbuf[cur][j]);
        }
      }
    }
    __syncthreads();   // all waves done with hbuf[cur] before it is reused
  }
}

__global__ void zero_kernel(float4* __restrict__ out, int n4) {
  int i      = blockIdx.x * blockDim.x + threadIdx.x;
  int stride = gridDim.x * blockDim.x;
  float4 z = make_float4(0.f, 0.f, 0.f, 0.f);
  for (; i < n4; i += stride) out[i] = z;
}

extern "C" void kernel_launch(void* const* d_in, const int* in_sizes, int n_in,
                              void* d_out, int out_size, void* d_ws, size_t ws_size,
                              hipStream_t stream) {
  const float*     H   = (const float*)d_in[0];
  const long long* Xn  = (const long long*)d_in[1];   // int64 indices
  float*           out = (float*)d_out;

  const int E      = in_sizes[1];
  const int nTiles = (E + TILE_EDGES - 1) / TILE_EDGES;

  // Output must be zeroed every call (atomics accumulate into it).
  const int n4 = out_size / 4;
  int zblocks = (n4 + 255) / 256;
  if (zblocks > 4096) zblocks = 4096;
  zero_kernel<<<zblocks, 256, 0, stream>>>((float4*)out, n4);

  int blocks = nTiles < 4096 ? nTiles : 4096;
  aggr_scatter_kernel<<<blocks, BLOCK, 0, stream>>>(H, Xn, out, E, nTiles);
}